// Graph2dConvolution_22230750724559
// MI455X (gfx1250) — compile-verified
//
#include <hip/hip_runtime.h>
#include <stdint.h>

// ---------------------------------------------------------------------------
// MI455X (gfx1250) implementation.
//   K1: 1x1-conv GEMM via V_WMMA_F32_16X16X4_F32, fused per-block LDS sums,
//       async global->LDS staging (GLOBAL_LOAD_ASYNC_TO_LDS_B128) when the
//       toolchain exposes the builtin.
//   K2: partial-sum reduction + block counts.
//   K3: per-batch graph math (means -> Y=means@W -> G=YY^T -> E -> adj_means).
//   K4: out = h + adj_means[b, idx[p], :]  (float4 streaming, LDS gather).
// ---------------------------------------------------------------------------

typedef __attribute__((ext_vector_type(2))) float v2f;
typedef __attribute__((ext_vector_type(8))) float v8f;
typedef int v4i __attribute__((vector_size(16)));   // builtin's b128 unit

#if defined(__has_builtin)
#if __has_builtin(__builtin_amdgcn_global_load_async_to_lds_b128) && \
    __has_builtin(__builtin_amdgcn_s_wait_asynccnt)
#define USE_ASYNC_LDS 1
#endif
#endif

// b128 async-copy operand casts (address-space casts were accepted by clang;
// the builtin wants int4-vector pointers).
#define AS1V(p) ((__attribute__((address_space(1))) v4i*)(p))
#define AS3V(p) ((__attribute__((address_space(3))) v4i*)(p))

// LDS layout strides (floats), padded for bank-conflict-free WMMA frag reads.
#define A_STRIDE 20    // 16 K-values + pad
#define X_STRIDE 132   // 128 pixels + pad (row start stays 16B aligned)

// ---------------------------------------------------------------------------
// K1: per-batch GEMM h[co,p] = sum_ci cw[co,ci] * x[b,ci,p]  (M=co, N=pixel)
//     + fused per-block (segment) partial sums.
// grid = B * 128 workgroups of 256 threads (8 waves); each WG: 128 pixels.
// LDS: As[256][20] + Xs[16][132] + Ss[64][256] + Bk[128]  = 94976 B
// ---------------------------------------------------------------------------
__global__ __launch_bounds__(256)
void conv1x1_wmma_kernel(const float* __restrict__ x,
                         const float* __restrict__ cw,
                         const int*   __restrict__ idx,
                         float*       __restrict__ h,
                         float*       __restrict__ partial)
{
    extern __shared__ float sm[];
    float* As = sm;                       // 256 * A_STRIDE
    float* Xs = As + 256 * A_STRIDE;      // 16  * X_STRIDE
    float* Ss = Xs + 16 * X_STRIDE;       // 64  * 256 block sums
    int*   Bk = (int*)(Ss + 64 * 256);    // 128 block ids

    const int t     = threadIdx.x;
    const int lane  = t & 31;
    const int wave  = t >> 5;             // 8 waves <-> 8 pixel N-tiles
    const int nIdx  = lane & 15;
    const int nHalf = lane >> 4;          // 0: K pair (0,1) | 1: K pair (2,3)
    const int b     = blockIdx.x >> 7;
    const int pg    = blockIdx.x & 127;
    const int p0    = pg << 7;            // 128 pixels per WG

    // zero LDS block-sum accumulators, load block ids for our pixels
    #pragma unroll
    for (int i = 0; i < 64; ++i) Ss[i * 256 + t] = 0.0f;
    if (t < 128) Bk[t] = idx[b * 16384 + p0 + t];

    v8f acc[16];
    #pragma unroll
    for (int mt = 0; mt < 16; ++mt) {
        #pragma unroll
        for (int r = 0; r < 8; ++r) acc[mt][r] = 0.0f;
    }

    const float* xb = x + ((size_t)b << 22);   // b * 256 * 16384

    for (int kc = 0; kc < 256; kc += 16) {
        __syncthreads();   // prior chunk fully consumed (also covers init)
#ifdef USE_ASYNC_LDS
        {
            // A chunk: row co=t, 16 consecutive K floats (64B) -> 4x b128
            const float* ga = cw + t * 256 + kc;
            float*       la = As + t * A_STRIDE;
            __builtin_amdgcn_global_load_async_to_lds_b128(AS1V(ga),      AS3V(la),      0, 0);
            __builtin_amdgcn_global_load_async_to_lds_b128(AS1V(ga + 4),  AS3V(la + 4),  0, 0);
            __builtin_amdgcn_global_load_async_to_lds_b128(AS1V(ga + 8),  AS3V(la + 8),  0, 0);
            __builtin_amdgcn_global_load_async_to_lds_b128(AS1V(ga + 12), AS3V(la + 12), 0, 0);
            // X chunk: 16 K-rows x 128 pixels -> 512 float4 slots, 2/thread
            #pragma unroll
            for (int i = 0; i < 2; ++i) {
                const int e  = i * 256 + t;
                const int r  = e >> 5;
                const int c4 = e & 31;
                const float* gx = xb + (size_t)(kc + r) * 16384 + p0 + c4 * 4;
                float*       lx = Xs + r * X_STRIDE + c4 * 4;
                __builtin_amdgcn_global_load_async_to_lds_b128(AS1V(gx), AS3V(lx), 0, 0);
            }
            __builtin_amdgcn_s_wait_asynccnt(0);
        }
#else
        {
            const float4* ga = (const float4*)(cw + t * 256 + kc);
            float4*       la = (float4*)(As + t * A_STRIDE);
            la[0] = ga[0]; la[1] = ga[1]; la[2] = ga[2]; la[3] = ga[3];
            #pragma unroll
            for (int i = 0; i < 2; ++i) {
                const int e  = i * 256 + t;
                const int r  = e >> 5;
                const int c4 = e & 31;
                *(float4*)(Xs + r * X_STRIDE + c4 * 4) =
                    *(const float4*)(xb + (size_t)(kc + r) * 16384 + p0 + c4 * 4);
            }
        }
#endif
        __syncthreads();   // staged data visible to all waves

        // 4 WMMA K-steps (K=4 each) over the 16-wide chunk
        #pragma unroll
        for (int ks = 0; ks < 4; ++ks) {
            const int ko = ks * 4 + nHalf * 2;
            // B fragment (4x16 of X): lane n holds X[ko..ko+1][ntile*16+n]
            v2f bf;
            bf.x = Xs[(ko + 0) * X_STRIDE + (wave << 4) + nIdx];
            bf.y = Xs[(ko + 1) * X_STRIDE + (wave << 4) + nIdx];
            #pragma unroll
            for (int mt = 0; mt < 16; ++mt) {
                // A fragment (16x4 of cw): lane m holds cw[m][ko..ko+1]
                const float* ap = As + (mt * 16 + nIdx) * A_STRIDE + ko;
                v2f af;
                af.x = ap[0];
                af.y = ap[1];
                acc[mt] = __builtin_amdgcn_wmma_f32_16x16x4_f32(
                    false, af, false, bf, (short)0, acc[mt], false, false);
            }
        }
    }

    // Epilogue: store h, LDS-accumulate per-block sums.
    // C/D layout: VGPR r -> rows (r, r+8); lanes 0-15 N=lane, 16-31 N=lane-16.
    const int    plocal = (wave << 4) + nIdx;
    const int    myblk  = Bk[plocal];
    const size_t pcol   = (size_t)p0 + plocal;
    #pragma unroll
    for (int mt = 0; mt < 16; ++mt) {
        #pragma unroll
        for (int r = 0; r < 8; ++r) {
            const int   co = mt * 16 + nHalf * 8 + r;
            const float v  = acc[mt][r];
            h[(((size_t)(b * 256 + co)) << 14) + pcol] = v;
            atomicAdd(&Ss[myblk * 256 + co], v);          // ds_add_f32
        }
    }
    __syncthreads();

    // Flush per-WG partial sums (plain coalesced stores, no global atomics)
    float* pd = partial + (size_t)blockIdx.x * 16384;
    #pragma unroll
    for (int i = 0; i < 64; ++i) pd[i * 256 + t] = Ss[i * 256 + t];
}

// ---------------------------------------------------------------------------
// K2a: sums[b,k,c] = sum over the batch's 128 per-WG partials
// ---------------------------------------------------------------------------
__global__ __launch_bounds__(256)
void reduce_partials_kernel(const float* __restrict__ partial,
                            float*       __restrict__ sums)
{
    const int gidx = blockIdx.x * 256 + threadIdx.x;   // 0 .. 16*64*256-1
    const int b    = gidx >> 14;
    const int rem  = gidx & 16383;
    const float* p = partial + ((size_t)b * 128) * 16384 + rem;
    float s = 0.0f;
    #pragma unroll 8
    for (int w = 0; w < 128; ++w) s += p[(size_t)w << 14];
    sums[gidx] = s;
}

// ---------------------------------------------------------------------------
// K2b: counts[b,k] (stored as float), one WG per batch
// ---------------------------------------------------------------------------
__global__ __launch_bounds__(256)
void counts_kernel(const int* __restrict__ idx, float* __restrict__ counts)
{
    __shared__ int cnt[64];
    const int b = blockIdx.x;
    if (threadIdx.x < 64) cnt[threadIdx.x] = 0;
    __syncthreads();
    for (int i = threadIdx.x; i < 16384; i += 256)
        atomicAdd(&cnt[idx[b * 16384 + i]], 1);
    __syncthreads();
    if (threadIdx.x < 64)
        counts[b * 64 + threadIdx.x] = (float)cnt[threadIdx.x];
}

// ---------------------------------------------------------------------------
// K3: per-batch graph math, one WG of 256 threads per batch (LDS resident).
//   means -> Y = means@W -> G = Y Y^T -> E_jk = exp(-(Gjj+Gkk-2Gjk)), diag 0
//   adjm[b,j,:] = sum_k E_jk * means[k,:]
// LDS: m[64*256] y[64*257] g[64*64] wc[64*256] cn[64] = 213504 B
// ---------------------------------------------------------------------------
__global__ __launch_bounds__(256)
void graph_adj_kernel(const float* __restrict__ sums,
                      const float* __restrict__ counts,
                      const float* __restrict__ Wm,
                      float*       __restrict__ adjm)
{
    extern __shared__ float sm[];
    float* m  = sm;               // 64 x 256
    float* y  = m + 64 * 256;     // 64 x 257 (pad -> conflict-free Y^T reads)
    float* g  = y + 64 * 257;     // 64 x 64
    float* wc = g + 64 * 64;      // 64 x 256 staged W chunk
    float* cn = wc + 64 * 256;    // 64

    const int t = threadIdx.x;
    const int b = blockIdx.x;

    if (t < 64) cn[t] = fmaxf(counts[b * 64 + t], 1.0f);
    __syncthreads();
    for (int i = 0; i < 64; ++i)
        m[i * 256 + t] = sums[(size_t)(b * 64 + i) * 256 + t] / cn[i];

    // Y = means @ W, W staged in 64-row chunks
    float acc[64];
    #pragma unroll
    for (int k = 0; k < 64; ++k) acc[k] = 0.0f;
    for (int cc = 0; cc < 256; cc += 64) {
        __syncthreads();
        for (int i = 0; i < 64; ++i)
            wc[i * 256 + t] = Wm[(size_t)(cc + i) * 256 + t];
        __syncthreads();
        for (int k = 0; k < 64; ++k) {
            float a = acc[k];
            for (int c = 0; c < 64; ++c)
                a = fmaf(m[k * 256 + cc + c], wc[c * 256 + t], a);
            acc[k] = a;
        }
    }
    #pragma unroll
    for (int k = 0; k < 64; ++k) y[k * 257 + t] = acc[k];
    __syncthreads();

    // G = Y Y^T (16 entries / thread; j uniform per 64-thread group)
    for (int i = 0; i < 16; ++i) {
        const int j  = i * 4 + (t >> 6);
        const int kk = t & 63;
        float s = 0.0f;
        for (int d = 0; d < 256; ++d)
            s = fmaf(y[j * 257 + d], y[kk * 257 + d], s);
        g[j * 64 + kk] = s;
    }
    __syncthreads();

    // E (in place over G; read everything first, then overwrite)
    float e[16];
    for (int i = 0; i < 16; ++i) {
        const int j  = i * 4 + (t >> 6);
        const int kk = t & 63;
        e[i] = (j == kk) ? 0.0f
             : __expf(-(g[j * 65] + g[kk * 65] - 2.0f * g[j * 64 + kk]));
    }
    __syncthreads();
    for (int i = 0; i < 16; ++i) {
        const int j  = i * 4 + (t >> 6);
        const int kk = t & 63;
        g[j * 64 + kk] = e[i];
    }
    __syncthreads();

    // adj_means = E @ means
    for (int j = 0; j < 64; ++j) {
        float s = 0.0f;
        for (int k = 0; k < 64; ++k)
            s = fmaf(g[j * 64 + k], m[k * 256 + t], s);
        adjm[((size_t)b * 64 + j) * 256 + t] = s;
    }
}

// ---------------------------------------------------------------------------
// K4: out[b,c,p] = h[b,c,p] + adjm[b, idx[b,p], c]
// grid = 16 batches * 16 pixel-groups; 256 threads * 4 pixels (float4).
// LDS holds adjm transposed [c][k] (stride 65) -> divergent k hits distinct
// banks, duplicate k broadcasts.
// ---------------------------------------------------------------------------
__global__ __launch_bounds__(256)
void gather_add_kernel(const float* __restrict__ h,
                       const int*   __restrict__ idx,
                       const float* __restrict__ adjm,
                       float*       __restrict__ out)
{
    extern __shared__ float am[];     // 256 * 65
    const int t  = threadIdx.x;
    const int b  = blockIdx.x >> 4;
    const int pg = blockIdx.x & 15;

    for (int i = 0; i < 64; ++i)
        am[t * 65 + i] = adjm[((size_t)b * 64 + i) * 256 + t];

    const int  p0 = pg * 1024 + t * 4;
    const int4 kv = *(const int4*)(idx + b * 16384 + p0);
    __syncthreads();

    const size_t basebp = ((size_t)b << 22) + (size_t)p0;
    #pragma unroll 4
    for (int c = 0; c < 256; ++c) {
        const size_t o = basebp + ((size_t)c << 14);
        float4 v = *(const float4*)(h + o);
        v.x += am[c * 65 + kv.x];
        v.y += am[c * 65 + kv.y];
        v.z += am[c * 65 + kv.z];
        v.w += am[c * 65 + kv.w];
        *(float4*)(out + o) = v;
    }
}

// ---------------------------------------------------------------------------
// Launch. Workspace (floats):
//   h        @ 0          : 67,108,864
//   partials @ 67,108,864 : 33,554,432
//   sums     @100,663,296 :    262,144
//   counts   @100,925,440 :      1,024
//   adjm     @100,926,464 :    262,144     (total ~405 MB)
// ---------------------------------------------------------------------------
extern "C" void kernel_launch(void* const* d_in, const int* in_sizes, int n_in,
                              void* d_out, int out_size, void* d_ws, size_t ws_size,
                              hipStream_t stream)
{
    (void)in_sizes; (void)n_in; (void)out_size; (void)ws_size;

    const float* x   = (const float*)d_in[0];
    const float* cw  = (const float*)d_in[1];
    const float* Wm  = (const float*)d_in[2];
    const int*   idx = (const int*)d_in[3];
    float*       out = (float*)d_out;
    float*       ws  = (float*)d_ws;

    float* h       = ws;
    float* partial = ws + 67108864ull;
    float* sums    = partial + 33554432ull;
    float* counts  = sums + 262144ull;
    float* adjm    = counts + 1024ull;

    const int shm_conv   = (256 * A_STRIDE + 16 * X_STRIDE + 64 * 256) * 4 + 128 * 4; // 94976
    const int shm_graph  = (64 * 256 + 64 * 257 + 64 * 64 + 64 * 256 + 64) * 4;       // 213504
    const int shm_gather = 256 * 65 * 4;                                              // 66560

    (void)hipFuncSetAttribute((const void*)conv1x1_wmma_kernel,
                              hipFuncAttributeMaxDynamicSharedMemorySize, shm_conv);
    (void)hipFuncSetAttribute((const void*)graph_adj_kernel,
                              hipFuncAttributeMaxDynamicSharedMemorySize, shm_graph);
    (void)hipFuncSetAttribute((const void*)gather_add_kernel,
                              hipFuncAttributeMaxDynamicSharedMemorySize, shm_gather);

    counts_kernel<<<dim3(16), dim3(256), 0, stream>>>(idx, counts);
    conv1x1_wmma_kernel<<<dim3(16 * 128), dim3(256), shm_conv, stream>>>(x, cw, idx, h, partial);
    reduce_partials_kernel<<<dim3(1024), dim3(256), 0, stream>>>(partial, sums);
    graph_adj_kernel<<<dim3(16), dim3(256), shm_graph, stream>>>(sums, counts, Wm, adjm);
    gather_add_kernel<<<dim3(16 * 16), dim3(256), shm_gather, stream>>>(h, idx, adjm, out);
}